// GraphAttentionLayer_8684423873295
// MI455X (gfx1250) — compile-verified
//
#include <hip/hip_runtime.h>
#include <math.h>

// GAT layer: b=2, t=4, N=256, F_in=128, F_out=64, leaky slope 0.2
// e[b,tau,i,j] = lrelu( s[b, tau', j] ),  tau' = (2*tau + (i>=128)) % 4
// s[b,tp,j]    = dot(h[b,tp,(2j)%256,:], a[0:64]) + dot(h[b,tp,(2j+1)%256,:], a[64:128])
// attn = softmax_j(adj * e);  out = attn @ h   (same tau)

typedef __attribute__((ext_vector_type(2))) float v2f;
typedef __attribute__((ext_vector_type(8))) float v8f;

#define BT    8      // b*t
#define NN    256
#define FIN   128
#define FOUT  64
#define LDSW  260    // padded LDS row stride (floats): conflict-free WMMA A reads

// ---------------- Kernel 1: h = x @ W  (f32 WMMA 16x16x4) ----------------
// grid = BT*16 blocks (bt, mtile), block = 128 threads (4 waves), wave w -> ntile w
__global__ __launch_bounds__(128) void k_gemm_xw(const float* __restrict__ x,
                                                 const float* __restrict__ W,
                                                 float* __restrict__ h) {
    const int blk   = blockIdx.x;
    const int bt    = blk >> 4;
    const int mtile = blk & 15;
    const int wv    = threadIdx.x >> 5;   // ntile 0..3
    const int lane  = threadIdx.x & 31;
    const int half  = lane >> 4;          // 0: K0..1 / rows 0..7,  1: K2..3 / rows 8..15
    const int l15   = lane & 15;

    const float* __restrict__ xrow = x + (size_t)(bt * NN + mtile * 16 + l15) * FIN;
    const int colbase = wv * 16 + l15;

    v8f c = {};
    for (int k0 = 0; k0 < FIN; k0 += 4) {
        const int ka = k0 + half * 2;
        v2f a_; a_.x = xrow[ka];                 a_.y = xrow[ka + 1];
        v2f b_; b_.x = W[ka * FOUT + colbase];   b_.y = W[(ka + 1) * FOUT + colbase];
        c = __builtin_amdgcn_wmma_f32_16x16x4_f32(false, a_, false, b_, (short)0, c,
                                                  false, false);
    }
    float* __restrict__ hout = h + (size_t)(bt * NN + mtile * 16) * FOUT + colbase;
#pragma unroll
    for (int r = 0; r < 8; ++r)
        hout[(size_t)(r + half * 8) * FOUT] = c[r];
}

// ---------------- Kernel 2: s[bt][j] ----------------
// grid = BT, block = 256 (thread j)
__global__ __launch_bounds__(256) void k_attn_s(const float* __restrict__ h,
                                                const float* __restrict__ a,
                                                float* __restrict__ s) {
    const int bt = blockIdx.x;
    const int j  = threadIdx.x;
    const float* __restrict__ hb = h + (size_t)bt * NN * FOUT;
    const int ja = (2 * j) & (NN - 1);
    const int jb = (2 * j + 1) & (NN - 1);
    const float* __restrict__ ra = hb + (size_t)ja * FOUT;
    const float* __restrict__ rb = hb + (size_t)jb * FOUT;
    float acc = 0.f;
#pragma unroll 8
    for (int f = 0; f < FOUT; ++f)
        acc = fmaf(ra[f], a[f], fmaf(rb[f], a[FOUT + f], acc));
    s[bt * NN + j] = acc;
}

// ---------------- Kernel 3: softmax(adj*lrelu(s)) then attn @ h ----------------
// grid = BT*16 blocks (bt, itile), block = 128 (4 waves)
__global__ __launch_bounds__(128) void k_attn_out(const float* __restrict__ adj,
                                                  const float* __restrict__ h,
                                                  const float* __restrict__ s,
                                                  float* __restrict__ out) {
    __shared__ float attn_lds[16][LDSW];

    const int blk   = blockIdx.x;
    const int bt    = blk >> 4;
    const int itile = blk & 15;
    const int b     = bt >> 2;
    const int tau   = bt & 3;
    const int cpred = (itile >= 8) ? 1 : 0;          // i >= 128
    const int tp    = (2 * tau + cpred) & 3;

    const int wv   = threadIdx.x >> 5;
    const int lane = threadIdx.x & 31;
    const int half = lane >> 4;
    const int l15  = lane & 15;

    const float* __restrict__ srow = s + (b * 4 + tp) * NN;
    const float* __restrict__ hb   = h + (size_t)bt * NN * FOUT;

    // ---- Phase 1: per-row softmax of adj*lrelu(s) into LDS (wave wv: 4 rows) ----
#pragma unroll
    for (int rr = 0; rr < 4; ++rr) {
        const int rloc = wv * 4 + rr;
        const int i    = itile * 16 + rloc;
        const float* __restrict__ arow = adj + ((size_t)(bt * NN + i)) * NN;

        float z[8];
        float mx = -3.402823466e38f;
#pragma unroll
        for (int u = 0; u < 8; ++u) {
            const int j = lane + u * 32;
            const float sv = srow[j];
            const float ev = (sv > 0.f) ? sv : 0.2f * sv;
            const float zz = arow[j] * ev;
            z[u] = zz;
            mx = fmaxf(mx, zz);
        }
#pragma unroll
        for (int off = 16; off > 0; off >>= 1)
            mx = fmaxf(mx, __shfl_xor(mx, off, 32));
        float sum = 0.f;
#pragma unroll
        for (int u = 0; u < 8; ++u) {
            const float p = __expf(z[u] - mx);
            z[u] = p;
            sum += p;
        }
#pragma unroll
        for (int off = 16; off > 0; off >>= 1)
            sum += __shfl_xor(sum, off, 32);
        const float inv = 1.f / sum;
#pragma unroll
        for (int u = 0; u < 8; ++u)
            attn_lds[rloc][lane + u * 32] = z[u] * inv;
    }
    __syncthreads();

    // ---- Phase 2: out tile = attn(16xN) @ h(NxFOUT); wave wv owns 16 cols ----
    const int colbase = wv * 16 + l15;
    v8f c = {};
    for (int k0 = 0; k0 < NN; k0 += 4) {
        const int ka = k0 + half * 2;
        v2f a_; a_.x = attn_lds[l15][ka];
                a_.y = attn_lds[l15][ka + 1];
        v2f b_; b_.x = hb[(size_t)ka * FOUT + colbase];
                b_.y = hb[(size_t)(ka + 1) * FOUT + colbase];
        c = __builtin_amdgcn_wmma_f32_16x16x4_f32(false, a_, false, b_, (short)0, c,
                                                  false, false);
    }
    float* __restrict__ orow = out + (size_t)(bt * NN + itile * 16) * FOUT + colbase;
#pragma unroll
    for (int r = 0; r < 8; ++r)
        orow[(size_t)(r + half * 8) * FOUT] = c[r];
}

extern "C" void kernel_launch(void* const* d_in, const int* in_sizes, int n_in,
                              void* d_out, int out_size, void* d_ws, size_t ws_size,
                              hipStream_t stream) {
    const float* x   = (const float*)d_in[0];   // [2,4,256,128]
    const float* adj = (const float*)d_in[1];   // [2,4,256,256]
    const float* W   = (const float*)d_in[2];   // [128,64]
    const float* a   = (const float*)d_in[3];   // [128,1]
    float* out = (float*)d_out;                 // [2,4,256,64]

    float* h = (float*)d_ws;                    // BT*NN*FOUT floats (512 KB)
    float* s = h + (size_t)BT * NN * FOUT;      // BT*NN floats (8 KB)

    k_gemm_xw<<<dim3(BT * 16), dim3(128), 0, stream>>>(x, W, h);
    k_attn_s<<<dim3(BT), dim3(256), 0, stream>>>(h, a, s);
    k_attn_out<<<dim3(BT * 16), dim3(128), 0, stream>>>(adj, h, s, out);
}